// GMMDense_17918603559390
// MI455X (gfx1250) — compile-verified
//
#include <hip/hip_runtime.h>
#include <hip/hip_bf16.h>

typedef __attribute__((ext_vector_type(2))) float v2f;
typedef __attribute__((ext_vector_type(8))) float v8f;

#define WAVES 4                    // waves per workgroup (wave32)
#define A_FLOATS (32 * 8 * 2 * 64) // 32 classes * 8 k-steps * 2 col-tiles * 64 floats = 32768

// ---------------------------------------------------------------------------
// Prep: per class k (one wave each), invert max(Sigma_k,0) with Gauss-Jordan
// + partial pivoting, then emit:
//   Afrag : A^-1 in WMMA B-fragment order (ds_load_b64-ready, conflict-free)
//   W[d][k] = ((A + A^T) mu_k)[d],   Cc[k] = mu_k^T A mu_k
// ---------------------------------------------------------------------------
__global__ void __launch_bounds__(32) gmm_prep(const float* __restrict__ Sigma,
                                               const float* __restrict__ Mu,
                                               float* __restrict__ Afrag,
                                               float* __restrict__ W,
                                               float* __restrict__ Cc) {
    __shared__ float M[32][33];
    __shared__ float Inv[32][33];
    __shared__ float red[32];
    __shared__ int   pivr;

    const int k = blockIdx.x;
    const int t = threadIdx.x;          // 0..31
    const float* S = Sigma + (size_t)k * 32 * 32;

    for (int j = 0; j < 32; ++j) {
        M[t][j]   = fmaxf(S[t * 32 + j], 0.0f);
        Inv[t][j] = (t == j) ? 1.0f : 0.0f;
    }
    __syncthreads();

    for (int i = 0; i < 32; ++i) {
        red[t] = (t >= i) ? fabsf(M[t][i]) : -1.0f;
        __syncthreads();
        if (t == 0) {
            int best = i; float bv = red[i];
            for (int r = i + 1; r < 32; ++r)
                if (red[r] > bv) { bv = red[r]; best = r; }
            pivr = best;
        }
        __syncthreads();
        const int p = pivr;
        if (p != i) {                    // swap rows i<->p, thread t = column t
            float tmp = M[i][t];  M[i][t]  = M[p][t];  M[p][t]  = tmp;
            tmp = Inv[i][t];      Inv[i][t] = Inv[p][t]; Inv[p][t] = tmp;
        }
        __syncthreads();
        const float pivinv = 1.0f / M[i][i];
        M[i][t]   *= pivinv;
        Inv[i][t] *= pivinv;
        __syncthreads();
        if (t != i) {                    // eliminate: thread t owns row t
            const float f = M[t][i];
            for (int j = 0; j < 32; ++j) {
                M[t][j]   = fmaf(-f, M[i][j],   M[t][j]);
                Inv[t][j] = fmaf(-f, Inv[i][j], Inv[t][j]);
            }
        }
        __syncthreads();
    }

    // Linear/constant terms
    const float* mu = Mu + k * 32;
    float wd = 0.0f, td = 0.0f;
    for (int e = 0; e < 32; ++e) {
        const float me = mu[e];
        wd = fmaf(Inv[t][e] + Inv[e][t], me, wd);
        td = fmaf(Inv[t][e], me, td);
    }
    W[t * 32 + k] = wd;                  // W layout: [d][k]
    red[t] = mu[t] * td;
    __syncthreads();
    if (t == 0) {
        float c = 0.0f;
        for (int d = 0; d < 32; ++d) c += red[d];
        Cc[k] = c;
    }

    // A^-1 in B-fragment order: record (k,s,tt), lane t holds
    //   { A[4s+2*hi][16tt+lm], A[4s+2*hi+1][16tt+lm] }
    const int hi = t >> 4, lm = t & 15;
    v2f* Af = (v2f*)Afrag;
    for (int s = 0; s < 8; ++s) {
        const int d0 = 4 * s + 2 * hi;
        for (int tt = 0; tt < 2; ++tt) {
            v2f v;
            v.x = Inv[d0][16 * tt + lm];
            v.y = Inv[d0 + 1][16 * tt + lm];
            Af[((k * 8 + s) * 2 + tt) * 32 + t] = v;
        }
    }
}

// ---------------------------------------------------------------------------
// Main: each wave owns a 16-row tile of X across all K=32 classes.
//   q[n,k] = x^T A_k x - x.w_k + c_k   via v_wmma_f32_16x16x4_f32
//   out    = exp(-0.5 max(q,0)) row-normalized (sum clamped at 1e-5)
// ---------------------------------------------------------------------------
__global__ void __launch_bounds__(32 * WAVES) gmm_main(const float* __restrict__ x,
                                                       const float* __restrict__ Afrag,
                                                       const float* __restrict__ W,
                                                       const float* __restrict__ Cc,
                                                       float* __restrict__ out,
                                                       int Nn) {
    __shared__ alignas(16) float lA[A_FLOATS];     // 128 KB: all 32 inverses, frag order
    __shared__ float ltile[WAVES][2][16][32];      // [wave][0=q,1=xw][row][k] : 16 KB

    const int tid  = threadIdx.x;
    const int wave = tid >> 5;
    const int lane = tid & 31;
    const int hi   = lane >> 4;
    const int lm   = lane & 15;

    const long base = (long)(blockIdx.x * WAVES + wave) * 16;

    // --- per-wave X loads first: overlap with the LDS fill below -----------
    v2f xa[8];                                      // A-operand frags (16x4 f32)
    {
        long row = base + lm; if (row >= Nn) row = Nn - 1;
        const v2f* x2 = (const v2f*)(x + row * 32);
        #pragma unroll
        for (int s = 0; s < 8; ++s) xa[s] = x2[2 * s + hi];   // K = 4s+2*hi, +1
    }
    float xc[2][8];                                 // X in C/D layout for row-dot
    #pragma unroll
    for (int r = 0; r < 8; ++r) {
        long row2 = base + r + 8 * hi; if (row2 >= Nn) row2 = Nn - 1;
        #pragma unroll
        for (int t = 0; t < 2; ++t) xc[t][r] = x[row2 * 32 + 16 * t + lm];
    }

    // --- linear term XW = X @ W (independent of LDS-A, before barrier) -----
    {
        v8f xw0 = (v8f)0.0f, xw1 = (v8f)0.0f;
        #pragma unroll
        for (int s = 0; s < 8; ++s) {
            const int d0 = 4 * s + 2 * hi;
            v2f b0, b1;
            b0.x = W[d0 * 32 + lm];        b0.y = W[(d0 + 1) * 32 + lm];
            b1.x = W[d0 * 32 + 16 + lm];   b1.y = W[(d0 + 1) * 32 + 16 + lm];
            xw0 = __builtin_amdgcn_wmma_f32_16x16x4_f32(false, xa[s], false, b0,
                                                        (short)0, xw0, false, false);
            xw1 = __builtin_amdgcn_wmma_f32_16x16x4_f32(false, xa[s], false, b1,
                                                        (short)0, xw1, false, false);
        }
        #pragma unroll
        for (int r = 0; r < 8; ++r) {
            ltile[wave][1][r + 8 * hi][lm]      = xw0[r];
            ltile[wave][1][r + 8 * hi][16 + lm] = xw1[r];
        }
    }

    // --- cooperative fill of fragment-ordered A^-1 into LDS ----------------
    {
        const float4* src = (const float4*)Afrag;
        float4*       dst = (float4*)lA;
        for (int i = tid; i < A_FLOATS / 4; i += 32 * WAVES) dst[i] = src[i];
    }
    __syncthreads();

    // --- main loop: Y = X * A_k (16 WMMA), rowdot(Y, X), per class ---------
    #pragma unroll 1
    for (int k = 0; k < 32; ++k) {
        const v2f* Af = ((const v2f*)lA) + k * 8 * 2 * 32;
        v8f acc0 = (v8f)0.0f, acc1 = (v8f)0.0f;
        #pragma unroll
        for (int s = 0; s < 8; ++s) {
            const v2f b0 = Af[(s * 2 + 0) * 32 + lane];   // one ds_load_b64 each,
            const v2f b1 = Af[(s * 2 + 1) * 32 + lane];   // bank-conflict-free
            acc0 = __builtin_amdgcn_wmma_f32_16x16x4_f32(false, xa[s], false, b0,
                                                         (short)0, acc0, false, false);
            acc1 = __builtin_amdgcn_wmma_f32_16x16x4_f32(false, xa[s], false, b1,
                                                         (short)0, acc1, false, false);
        }
        #pragma unroll
        for (int r = 0; r < 8; ++r) {
            float p = acc0[r] * xc[0][r] + acc1[r] * xc[1][r];
            p += __shfl_xor(p, 8, 32);       // reduce within 16-lane halves
            p += __shfl_xor(p, 4, 32);
            p += __shfl_xor(p, 2, 32);
            p += __shfl_xor(p, 1, 32);
            if (lm == 0) ltile[wave][0][r + 8 * hi][k] = p;   // lanes 0 & 16
        }
    }

    // --- epilogue: lane = class k; exp, row-sum over K, normalize ----------
    const int kk = lane;
    const float ck = Cc[kk];
    #pragma unroll 1
    for (int r = 0; r < 16; ++r) {
        const long row = base + r;
        float q = ltile[wave][0][r][kk] - ltile[wave][1][r][kk] + ck;
        q = fmaxf(q, 0.0f);
        const float e = expf(-0.5f * q);
        float s = e;
        s += __shfl_xor(s, 16, 32);
        s += __shfl_xor(s, 8, 32);
        s += __shfl_xor(s, 4, 32);
        s += __shfl_xor(s, 2, 32);
        s += __shfl_xor(s, 1, 32);
        s = fmaxf(s, 1e-5f);
        if (row < Nn) out[row * 32 + kk] = e / s;
    }
}

// ---------------------------------------------------------------------------
extern "C" void kernel_launch(void* const* d_in, const int* in_sizes, int n_in,
                              void* d_out, int out_size, void* d_ws, size_t ws_size,
                              hipStream_t stream) {
    (void)n_in; (void)out_size; (void)ws_size;
    const float* x     = (const float*)d_in[0];
    // d_in[1] = prob : not used by the forward output
    const float* Mu    = (const float*)d_in[2];
    const float* Sigma = (const float*)d_in[3];
    float* out  = (float*)d_out;

    float* Afrag = (float*)d_ws;             // 32768 floats, fragment order
    float* W     = Afrag + A_FLOATS;         // 1024 floats  [d][k]
    float* Cc    = W + 32 * 32;              // 32 floats

    const int N = in_sizes[0] / 32;

    gmm_prep<<<32, 32, 0, stream>>>(Sigma, Mu, Afrag, W, Cc);

    const int rows_per_block = 16 * WAVES;
    const int blocks = (N + rows_per_block - 1) / rows_per_block;
    gmm_main<<<blocks, 32 * WAVES, 0, stream>>>(x, Afrag, W, Cc, out, N);
}